// LinearCE_69982197121081
// MI455X (gfx1250) — compile-verified
//
#include <hip/hip_runtime.h>
#include <math.h>

typedef __attribute__((ext_vector_type(2))) float v2f;
typedef __attribute__((ext_vector_type(8))) float v8f;

#define BATCH 8192
#define DDIM  512
#define VDIM  50257

#define MT 128               // block tile M
#define NT 128               // block tile N
#define KC 32                // K chunk staged in LDS
#define XS_STRIDE 36         // floats; 144B row stride: 16B-aligned b128 writes,
                             // 8B-aligned b64 reads, conflict-free compute reads

// Async DMA: global -> LDS, no VGPR round-trip (ASYNCcnt-tracked).
__device__ __forceinline__ void async_ld_b128(unsigned lds_addr, const float* g) {
    asm volatile("global_load_async_to_lds_b128 %0, %1, off"
                 :: "v"(lds_addr), "v"((unsigned long long)(size_t)g)
                 : "memory");
}
__device__ __forceinline__ void wait_async0() {
    asm volatile("s_wait_asynccnt 0x0" ::: "memory");
}

// ---------------------------------------------------------------------------
// Kernel 1: logits = x @ W via V_WMMA_F32_16X16X4_F32 (fp32 in, fp32 acc)
// grid = (ceil(V/128), 8192/128), block = 256 (8 waves).
// Wave w computes a 128(M) x 16(N) slab: 8 row-tiles share each B fragment.
// x tile double-buffered in LDS via async global->LDS DMA; W streams from the
// 192 MB L2 (103 MB resident) with immediate-offset b32 loads.
// ---------------------------------------------------------------------------
__global__ __launch_bounds__(256) void lince_gemm_kernel(
    const float* __restrict__ x, const float* __restrict__ W,
    float* __restrict__ out)
{
    __shared__ float xs[2][MT * XS_STRIDE];

    const int tid  = threadIdx.x;
    const int wave = tid >> 5;        // 0..7 -> 16-col slab
    const int lane = tid & 31;
    const int half = lane >> 4;       // lane half: K pair (A/B), M+8 (C/D)
    const int l16  = lane & 15;

    const int rowBase = blockIdx.y * MT;
    const int colBase = blockIdx.x * NT;
    const int col     = colBase + wave * 16 + l16;     // B/C/D: N = lane&15
    const bool colOk  = (col < VDIM);
    const int colC    = colOk ? col : (VDIM - 1);      // clamped address

    // Staging geometry: 1024 float4 per tile, 4 per thread, coalesced b128.
    int sRow[4], sCol[4];
    unsigned ldsOff[2][4];
    #pragma unroll
    for (int t = 0; t < 4; ++t) {
        const int idx = tid + t * 256;   // float4 index in 128x32 tile
        sRow[t] = idx >> 3;              // 8 float4 per row
        sCol[t] = (idx & 7) * 4;         // float offset in row
        ldsOff[0][t] = (unsigned)(size_t)&xs[0][sRow[t] * XS_STRIDE + sCol[t]];
        ldsOff[1][t] = (unsigned)(size_t)&xs[1][sRow[t] * XS_STRIDE + sCol[t]];
    }

    v8f acc[8];
    #pragma unroll
    for (int r = 0; r < 8; ++r) acc[r] = (v8f){};

    // Prologue: DMA chunk 0 into buffer 0.
    #pragma unroll
    for (int t = 0; t < 4; ++t)
        async_ld_b128(ldsOff[0][t],
                      &x[(size_t)(rowBase + sRow[t]) * DDIM + sCol[t]]);
    wait_async0();
    __syncthreads();

    int buf = 0;
    for (int kc = 0; kc < DDIM; kc += KC, buf ^= 1) {
        const bool hasNext = (kc + KC) < DDIM;

        // Issue next chunk's DMA early; it overlaps the 64 WMMAs below.
        // Safe: buf^1 was last read before the barrier that ended the
        // previous iteration.
        if (hasNext) {
            #pragma unroll
            for (int t = 0; t < 4; ++t)
                async_ld_b128(ldsOff[buf ^ 1][t],
                              &x[(size_t)(rowBase + sRow[t]) * DDIM
                                 + (kc + KC) + sCol[t]]);
        }

        // Per-chunk loop-invariant bases -> immediate-offset loads inside
        // (max W offset 29*V*4 ~ 5.8 MB < signed 24-bit IOFFSET).
        const float* wp = W + (size_t)(kc + 2 * half) * VDIM + colC;
        const float* xp = &xs[buf][l16 * XS_STRIDE + 2 * half];

        #pragma unroll
        for (int kk = 0; kk < KC; kk += 4) {
            // B fragment (4x16): VGPR0 = K(+2*half), VGPR1 = K+1, N = l16.
            const float b0 = wp[(size_t)kk * VDIM];
            const float b1 = wp[(size_t)(kk + 1) * VDIM];
            v2f b;
            b.x = colOk ? b0 : 0.0f;
            b.y = colOk ? b1 : 0.0f;

            #pragma unroll
            for (int r = 0; r < 8; ++r) {
                // A fragment (16x4): row = 16r + l16, K pair by lane half.
                const v2f a = *(const v2f*)&xp[16 * r * XS_STRIDE + kk];
                acc[r] = __builtin_amdgcn_wmma_f32_16x16x4_f32(
                    false, a, false, b, (short)0, acc[r], false, false);
            }
        }

        if (hasNext) {
            wait_async0();      // this wave's DMA into buf^1 complete
            __syncthreads();    // all waves' DMA visible; buf reads sealed
        }
    }

    if (colOk) {
        #pragma unroll
        for (int r = 0; r < 8; ++r) {
            const int row0 = rowBase + 16 * r + 8 * half;  // lanes>=16 -> M+8
            size_t base = (size_t)row0 * VDIM + col;
            #pragma unroll
            for (int v = 0; v < 8; ++v)                    // VGPR v -> M = v
                out[base + (size_t)v * VDIM] = acc[r][v];
        }
    }
}

// ---------------------------------------------------------------------------
// Kernel 2: per-row online softmax stats + NLL.  One 256-thread block per row.
// ---------------------------------------------------------------------------
__global__ __launch_bounds__(256) void lince_rowsoftmax_kernel(
    const float* __restrict__ logits, const int* __restrict__ label,
    float* __restrict__ nll)
{
    const int row = blockIdx.x;
    const float* lrow = logits + (size_t)row * VDIM;

    float m = -INFINITY, s = 0.0f;
    for (int j = threadIdx.x; j < VDIM; j += 256) {
        const float v  = lrow[j];
        const float nm = fmaxf(m, v);
        s = s * __expf(m - nm) + __expf(v - nm);
        m = nm;
    }

    __shared__ float sm[256], ss[256];
    sm[threadIdx.x] = m;
    ss[threadIdx.x] = s;
    __syncthreads();
    for (int off = 128; off > 0; off >>= 1) {
        if (threadIdx.x < off) {
            const float m1 = sm[threadIdx.x], s1 = ss[threadIdx.x];
            const float m2 = sm[threadIdx.x + off], s2 = ss[threadIdx.x + off];
            const float nm = fmaxf(m1, m2);
            ss[threadIdx.x] = s1 * __expf(m1 - nm) + s2 * __expf(m2 - nm);
            sm[threadIdx.x] = nm;
        }
        __syncthreads();
    }

    if (threadIdx.x == 0) {
        const int lbl = label[row];
        nll[row] = (sm[0] + __logf(ss[0])) - lrow[lbl];
    }
}

// ---------------------------------------------------------------------------
// Kernel 3: mean of 8192 NLLs -> scalar loss at d_out[BATCH*V]
// ---------------------------------------------------------------------------
__global__ __launch_bounds__(256) void lince_meanloss_kernel(
    const float* __restrict__ nll, float* __restrict__ lossOut)
{
    __shared__ float red[256];
    float s = 0.0f;
    for (int i = threadIdx.x; i < BATCH; i += 256) s += nll[i];
    red[threadIdx.x] = s;
    __syncthreads();
    for (int off = 128; off > 0; off >>= 1) {
        if (threadIdx.x < off) red[threadIdx.x] += red[threadIdx.x + off];
        __syncthreads();
    }
    if (threadIdx.x == 0) *lossOut = red[0] / (float)BATCH;
}

extern "C" void kernel_launch(void* const* d_in, const int* in_sizes, int n_in,
                              void* d_out, int out_size, void* d_ws, size_t ws_size,
                              hipStream_t stream) {
    (void)in_sizes; (void)n_in; (void)out_size; (void)ws_size;
    const float* x     = (const float*)d_in[0];
    const int*   label = (const int*)d_in[1];
    const float* W     = (const float*)d_in[2];
    float* out = (float*)d_out;
    float* nll = (float*)d_ws;          // 8192 floats = 32 KB scratch

    dim3 grid((VDIM + NT - 1) / NT, BATCH / MT);   // (393, 64)
    lince_gemm_kernel<<<grid, 256, 0, stream>>>(x, W, out);
    lince_rowsoftmax_kernel<<<BATCH, 256, 0, stream>>>(out, label, nll);
    lince_meanloss_kernel<<<1, 256, 0, stream>>>(nll, out + (size_t)BATCH * VDIM);
}